// RelationAwareAttention_4561255268492
// MI455X (gfx1250) — compile-verified
//
#include <hip/hip_runtime.h>
#include <hip/hip_bf16.h>

// ---------------------------------------------------------------------------
// Relation-aware attention for MI455X (gfx1250), f32 end-to-end.
// Memory-bound (~150MB traffic vs ~13 GFLOP) -> keep f32, use
// V_WMMA_F32_16X16X4_F32 chained over K for all four matmuls.
// B-operand guards removed structurally (zero rows skipped), loads batched
// ahead of WMMA chains, Q tile staged via async-to-LDS, phase-6 work split
// into 24 equal partial-K tasks so all 8 waves carry identical load.
// ---------------------------------------------------------------------------

typedef __attribute__((ext_vector_type(2))) float v2f;
typedef __attribute__((ext_vector_type(8))) float v8f;

#define BS     8
#define NH     8
#define SEQ    512
#define HD     64
#define NP     1024      // P
#define NPAD   1040      // P+1 padded up to multiple of 16
#define TILE_M 16
#define PPITCH 520       // p-tile row pitch (512 + pad)

__device__ __forceinline__ v8f wmma_f32_k4(v2f a, v2f b, v8f c) {
    return __builtin_amdgcn_wmma_f32_16x16x4_f32(
        /*neg_a=*/false, a, /*neg_b=*/false, b,
        /*c_mod=*/(short)0, c, /*reuse_a=*/false, /*reuse_b=*/false);
}

__global__ __launch_bounds__(256)
void relattn_kernel(const float* __restrict__ Q,
                    const float* __restrict__ K,
                    const float* __restrict__ V,
                    const float* __restrict__ RK,
                    const float* __restrict__ RV,
                    const long long* __restrict__ PM,   // int64 path_map [bs,L,L]
                    const int* __restrict__ MASK,       // int32 [bs,1,L,L]
                    float* __restrict__ OUT_attn,       // [bs,h,L,d]
                    float* __restrict__ OUT_p)          // [bs,h,L,L]
{
    __shared__ float SA[TILE_M * NPAD];    // score_all tile, then scatter tile (66.5 KB)
    __shared__ float PT[TILE_M * PPITCH];  // p_attn tile (33.3 KB)
    __shared__ float QS[TILE_M * HD];      // Q tile (4 KB)
    __shared__ float OT[TILE_M * HD];      // output tile (4 KB)

    const int bh    = blockIdx.y;          // 0..63  (b*NH + h)
    const int b     = bh >> 3;
    const int ib    = blockIdx.x * TILE_M; // i-tile base row
    const int tid   = threadIdx.x;
    const int lane  = tid & 31;
    const int wave  = tid >> 5;            // 0..7
    const int lrow  = lane & 15;           // M/N index within 16
    const int lhalf = lane >> 4;           // 0/1 : K low/high pair

    const float* Qbh  = Q  + (size_t)bh * SEQ * HD;
    const float* Kbh  = K  + (size_t)bh * SEQ * HD;
    const float* Vbh  = V  + (size_t)bh * SEQ * HD;
    const float* RKbh = RK + (size_t)bh * NP * HD;
    const float* RVbh = RV + (size_t)bh * NP * HD;
    const long long* PMb = PM   + (size_t)b * SEQ * SEQ + (size_t)ib * SEQ;
    const int*       Mb  = MASK + (size_t)b * SEQ * SEQ + (size_t)ib * SEQ;

    // ---- Phase 0: stage Q tile via gfx1250 async-to-LDS (16B/lane = 4KB) ---
    {
        // LDS byte offset = low 32 bits of the generic shared address
        // (passed to the instruction as its LDS address operand).
        unsigned lds_off = (unsigned)(unsigned long long)(uintptr_t)(&QS[tid * 4]);
        unsigned long long gaddr =
            (unsigned long long)(uintptr_t)(Qbh + (size_t)ib * HD + tid * 4);
        asm volatile("global_load_async_to_lds_b128 %0, %1, off"
                     :: "v"(lds_off), "v"(gaddr) : "memory");
    }
    // Warm the gather tables behind the matmul phases:
    // path_map tile = 16*512*8B = 64KB (512 lines), mask tile = 32KB (256 lines).
    __builtin_prefetch(PMb + (size_t)(tid * 2 + 0) * 16, 0, 1);
    __builtin_prefetch(PMb + (size_t)(tid * 2 + 1) * 16, 0, 1);
    __builtin_prefetch(Mb  + (size_t)tid * 32,           0, 1);
    asm volatile("s_wait_asynccnt 0" ::: "memory");
    __syncthreads();

    // 32-bit A 16x4 layout: lanes 0-15 -> K pair {0,1}, lanes 16-31 -> {2,3}
    v2f qa[16];
    #pragma unroll
    for (int kc = 0; kc < 16; ++kc)
        qa[kc] = *(const v2f*)&QS[lrow * HD + kc * 4 + lhalf * 2];

    // ---- Phase 1: score_all tile = Q_tile @ r_k_ext^T ----------------------
    // rows >= NP of r_k_ext are zero -> only the 64 real n-tiles computed;
    // SA columns 1024..1039 are zero-filled (gather only reads idx <= 1024).
    for (int e = tid; e < TILE_M * 16; e += 256)
        SA[(e >> 4) * NPAD + NP + (e & 15)] = 0.0f;

    for (int nt = wave; nt < NP / 16; nt += 8) {
        const int nb = nt * 16;
        const float* rrow = RKbh + (size_t)(nb + lrow) * HD + lhalf * 2;
        v2f bf[16];
        #pragma unroll
        for (int kc = 0; kc < 16; ++kc)
            bf[kc] = *(const v2f*)(rrow + kc * 4);       // global_load_b64, batched
        v8f acc = {};
        #pragma unroll
        for (int kc = 0; kc < 16; ++kc)
            acc = wmma_f32_k4(qa[kc], bf[kc], acc);
        #pragma unroll
        for (int r = 0; r < 8; ++r) {                    // C/D: vgpr r -> M = r + 8*half
            const int m = r + lhalf * 8;
            SA[m * NPAD + nb + lrow] = acc[r];
        }
    }
    __syncthreads();

    // ---- Phase 2: S = Q K^T + gathered bias, scale, mask -------------------
    for (int jt = wave; jt < SEQ / 16; jt += 8) {
        const int jb = jt * 16;
        const float* krow = Kbh + (size_t)(jb + lrow) * HD + lhalf * 2;
        v2f bf[16];
        #pragma unroll
        for (int kc = 0; kc < 16; ++kc)
            bf[kc] = *(const v2f*)(krow + kc * 4);
        v8f acc = {};
        #pragma unroll
        for (int kc = 0; kc < 16; ++kc)
            acc = wmma_f32_k4(qa[kc], bf[kc], acc);
        #pragma unroll
        for (int r = 0; r < 8; ++r) {
            const int i = r + lhalf * 8;
            const int j = jb + lrow;
            const int idx = (int)PMb[(size_t)i * SEQ + j];    // 0..1024
            float s = (acc[r] + SA[i * NPAD + idx]) * 0.125f; // 1/sqrt(64)
            if (Mb[(size_t)i * SEQ + j] == 0) s = -1e9f;
            PT[i * PPITCH + j] = s;
        }
    }
    __syncthreads();

    // ---- Phase 3: zero scatter target + output tile ------------------------
    for (int e = tid; e < TILE_M * NPAD; e += 256) SA[e] = 0.0f;
    for (int e = tid; e < TILE_M * HD;   e += 256) OT[e] = 0.0f;

    // ---- Phase 4: row softmax (16 lanes per row, shfl width 16) -----------
    {
        const int row = tid >> 4;                // 0..15
        const int c0  = tid & 15;
        float mx = -3.4e38f;
        for (int c = c0; c < SEQ; c += 16)
            mx = fmaxf(mx, PT[row * PPITCH + c]);
        #pragma unroll
        for (int m = 1; m < 16; m <<= 1)
            mx = fmaxf(mx, __shfl_xor(mx, m, 16));
        float sum = 0.0f;
        for (int c = c0; c < SEQ; c += 16) {
            const float e = __expf(PT[row * PPITCH + c] - mx);
            PT[row * PPITCH + c] = e;
            sum += e;
        }
        #pragma unroll
        for (int m = 1; m < 16; m <<= 1)
            sum += __shfl_xor(sum, m, 16);
        const float inv = 1.0f / sum;
        for (int c = c0; c < SEQ; c += 16)
            PT[row * PPITCH + c] *= inv;
    }
    __syncthreads();

    // ---- Phase 5: write p_attn out; scatter path mass into SA (ds_add_f32)
    for (int e = tid; e < TILE_M * SEQ; e += 256) {
        const int i = e >> 9;
        const int j = e & 511;
        const float p = PT[i * PPITCH + j];
        OUT_p[((size_t)bh * SEQ + ib + i) * SEQ + j] = p;
        const int idx = (int)PMb[(size_t)i * SEQ + j];
        atomicAdd(&SA[i * NPAD + idx], p);
    }
    __syncthreads();

    // ---- Phase 6: O = P@V + scat@r_v_ext, 24 equal 64-chunk tasks ----------
    // P@V     : K=512  -> 4 n-tiles x 2 K-halves   = 8 tasks  (1 per wave)
    // scat@r_v: K=1024 -> 4 n-tiles x 4 K-quarters = 16 tasks (2 per wave)
    // Every wave runs exactly 192 chunks; partial sums merge via ds atomics.
    {
        // --- one P@V half-tile per wave ---
        const int col = (wave >> 1) * 16 + lrow;
        const int kc0 = (wave & 1) * 64;
        v8f acc = {};
        for (int kb = kc0; kb < kc0 + 64; kb += 4) {
            v2f a[4], bf[4];
            #pragma unroll
            for (int u = 0; u < 4; ++u) {
                const int kk = (kb + u) * 4 + lhalf * 2;
                a[u]    = *(const v2f*)&PT[lrow * PPITCH + kk];
                bf[u].x = Vbh[(size_t)kk * HD + col];          // B non-transposed
                bf[u].y = Vbh[(size_t)(kk + 1) * HD + col];
            }
            #pragma unroll
            for (int u = 0; u < 4; ++u)
                acc = wmma_f32_k4(a[u], bf[u], acc);
        }
        #pragma unroll
        for (int r = 0; r < 8; ++r)
            atomicAdd(&OT[(r + lhalf * 8) * HD + col], acc[r]);
    }
    for (int t = wave; t < 16; t += 8) {
        // --- two scat@r_v quarter-tiles per wave ---
        // r_v_ext rows >= 1024 are zero, so K stops at 1024 (no guard).
        const int col = (t >> 2) * 16 + lrow;
        const int kc0 = (t & 3) * 64;
        v8f acc = {};
        for (int kb = kc0; kb < kc0 + 64; kb += 4) {
            v2f a[4], bf[4];
            #pragma unroll
            for (int u = 0; u < 4; ++u) {
                const int kk = (kb + u) * 4 + lhalf * 2;
                a[u]    = *(const v2f*)&SA[lrow * NPAD + kk];
                bf[u].x = RVbh[(size_t)kk * HD + col];
                bf[u].y = RVbh[(size_t)(kk + 1) * HD + col];
            }
            #pragma unroll
            for (int u = 0; u < 4; ++u)
                acc = wmma_f32_k4(a[u], bf[u], acc);
        }
        #pragma unroll
        for (int r = 0; r < 8; ++r)
            atomicAdd(&OT[(r + lhalf * 8) * HD + col], acc[r]);
    }
    __syncthreads();

    // ---- Phase 7: write attn_sum tile -------------------------------------
    for (int e = tid; e < TILE_M * HD; e += 256)
        OUT_attn[((size_t)bh * SEQ + ib) * HD + e] = OT[e];
}

extern "C" void kernel_launch(void* const* d_in, const int* in_sizes, int n_in,
                              void* d_out, int out_size, void* d_ws, size_t ws_size,
                              hipStream_t stream) {
    (void)in_sizes; (void)n_in; (void)out_size; (void)d_ws; (void)ws_size;

    const float*     Q    = (const float*)d_in[0];
    const float*     Kt   = (const float*)d_in[1];
    const float*     V    = (const float*)d_in[2];
    const float*     RK   = (const float*)d_in[3];
    const float*     RV   = (const float*)d_in[4];
    const long long* PM   = (const long long*)d_in[5];
    const int*       MASK = (const int*)d_in[6];

    float* OUT_attn = (float*)d_out;                                   // [bs,h,L,d]
    float* OUT_p    = (float*)d_out + (size_t)BS * NH * SEQ * HD;      // [bs,h,L,L]

    dim3 grid(SEQ / TILE_M, BS * NH);   // 32 x 64
    dim3 block(256);
    relattn_kernel<<<grid, block, 0, stream>>>(Q, Kt, V, RK, RV, PM, MASK,
                                               OUT_attn, OUT_p);
}